// ConditionalRandomField_60928406061116
// MI455X (gfx1250) — compile-verified
//
#include <hip/hip_runtime.h>
#include <hip/hip_bf16.h>
#include <math.h>

// Fixed problem shape from setup_inputs(): B=2, K=21, H=W=256.
#define KCLS   21
#define SROW   24          // padded row stride (floats) -> 6 WMMA K-steps of 4
#define HWPIX  65536       // H*W
#define HWSH   16          // log2(HWPIX)

typedef __attribute__((ext_vector_type(2))) float v2f;
typedef __attribute__((ext_vector_type(8))) float v8f;

// ---------------------------------------------------------------- helpers
__device__ __forceinline__ float maskval(int k, int j) {
    // mask = 1 - I ; mask[20,:] *= 10 ; mask[:,20] *= 10 ; zero padding beyond 21
    if (k >= KCLS || j >= KCLS) return 0.0f;
    if (k == j) return 0.0f;
    return (k == KCLS - 1 || j == KCLS - 1) ? 10.0f : 1.0f;
}

__device__ __forceinline__ void atomAddF(float* p, float v) {
    unsafeAtomicAdd(p, v);   // hardware global_atomic_add_f32 (no-return -> STOREcnt)
}

// ------------------------------------------------ 1) softmax(feat) -> Q0, unary
__global__ void crf_init_kernel(const float* __restrict__ feat,
                                float* __restrict__ Q,
                                float* __restrict__ unary, int N) {
    int n = blockIdx.x * blockDim.x + threadIdx.x;
    if (n >= N) return;
    int b = n >> HWSH, hw = n & (HWPIX - 1);
    const float* f = feat + ((size_t)b * KCLS) * HWPIX + hw;   // f[k*HWPIX]
    float x[KCLS];
    float m = -3.402823e38f;
#pragma unroll
    for (int k = 0; k < KCLS; ++k) { x[k] = f[(size_t)k * HWPIX]; m = fmaxf(m, x[k]); }
    float s = 0.0f;
#pragma unroll
    for (int k = 0; k < KCLS; ++k) s += expf(x[k] - m);
    float lse = m + logf(s);
    float* q = Q + (size_t)n * SROW;
    float* u = unary + (size_t)n * SROW;
#pragma unroll
    for (int k = 0; k < KCLS; ++k) { q[k] = expf(x[k] - lse); u[k] = lse - x[k]; }
    q[21] = q[22] = q[23] = 0.0f;
    u[21] = u[22] = u[23] = 0.0f;
}

// ------------------------------------------------ 2) per-edge kernel weights
__global__ void crf_edge_kern_kernel(const float* __restrict__ img,
                                     const float* __restrict__ dist_diff,
                                     const int*   __restrict__ msg_node,
                                     const float* __restrict__ alpha,  // 2
                                     const float* __restrict__ beta,   // 3
                                     const float* __restrict__ gamma,  // 2
                                     const float* __restrict__ w1,     // 1
                                     const float* __restrict__ w2,     // 1
                                     float* __restrict__ kern, int E) {
    int e = blockIdx.x * blockDim.x + threadIdx.x;
    if (e >= E) return;
    int nin  = msg_node[2 * e];
    int nout = msg_node[2 * e + 1];
    float d0 = dist_diff[2 * e], d1 = dist_diff[2 * e + 1];

    int bi = nin >> HWSH,  hi = nin  & (HWPIX - 1);
    int bo = nout >> HWSH, ho = nout & (HWPIX - 1);
    float it = 0.0f;
#pragma unroll
    for (int c = 0; c < 3; ++c) {
        float a = img[((size_t)bi * 3 + c) * HWPIX + hi];
        float b = img[((size_t)bo * 3 + c) * HWPIX + ho];
        float d = a - b;
        float bc = beta[c];
        it += (d * d) / (2.0f * bc * bc);
    }
    float a0 = alpha[0], a1 = alpha[1];
    float g0 = gamma[0], g1 = gamma[1];
    float at = d0 / (2.0f * a0 * a0) + d1 / (2.0f * a1 * a1);
    float gt = d0 / (2.0f * g0 * g0) + d1 / (2.0f * g1 * g1);
    kern[e] = fabsf(w1[0]) * expf(-at - it) + fabsf(w2[0]) * expf(-gt);
}

// ------------------------------------------------ 3) zero agg
__global__ void crf_zero_kernel(float4* __restrict__ p, int n4) {
    int i = blockIdx.x * blockDim.x + threadIdx.x;
    int stride = gridDim.x * blockDim.x;
    float4 z = make_float4(0.f, 0.f, 0.f, 0.f);
    for (; i < n4; i += stride) p[i] = z;
}

// ------------------------------------------------ 4) edge scatter: agg[out] += kern * Q[in]
__global__ void crf_scatter_kernel(const float* __restrict__ Q,
                                   const float* __restrict__ kern,
                                   const int*   __restrict__ msg_node,
                                   float* __restrict__ agg, int E) {
    int e = blockIdx.x * blockDim.x + threadIdx.x;
    if (e >= E) return;
    int nin  = msg_node[2 * e];
    int nout = msg_node[2 * e + 1];
    float kv = kern[e];
    const float4* q4 = (const float4*)(Q + (size_t)nin * SROW);   // 96B row, 16B aligned
    float4 v0 = q4[0], v1 = q4[1], v2 = q4[2], v3 = q4[3], v4 = q4[4], v5 = q4[5];
    float qv[SROW] = { v0.x, v0.y, v0.z, v0.w, v1.x, v1.y, v1.z, v1.w,
                       v2.x, v2.y, v2.z, v2.w, v3.x, v3.y, v3.z, v3.w,
                       v4.x, v4.y, v4.z, v4.w, v5.x, v5.y, v5.z, v5.w };
    float* ar = agg + (size_t)nout * SROW;
#pragma unroll
    for (int k = 0; k < KCLS; ++k) atomAddF(&ar[k], kv * qv[k]);
}

// ------------------------------------------------ 5) agg <- agg @ mask, via V_WMMA_F32_16X16X4_F32
// One wave owns one 16-row tile: 6 K-steps x 2 col-tiles = 12 WMMAs.
__global__ void crf_wmma_mask_kernel(float* __restrict__ agg, int N) {
    const int lane = threadIdx.x & 31;
    const int wave = threadIdx.x >> 5;
    const int wavesPerBlock = blockDim.x >> 5;
    const int tile = blockIdx.x * wavesPerBlock + wave;   // grid sized exactly: no tail
    const int p0 = tile * 16;

    const int hf = lane >> 4;      // half-wave: 0 or 1
    const int l  = lane & 15;

    v8f c0 = {};   // output cols 0..15
    v8f c1 = {};   // output cols 16..23 (+pad)

    const int rowA = p0 + l;       // A: M across lanes (both halves read all 16 rows)
#pragma unroll
    for (int s = 0; s < 6; ++s) {
        const int k0 = 4 * s + 2 * hf;   // A/B: K pair per lane-half
        v2f a;
        a.x = agg[rowA * SROW + k0];
        a.y = agg[rowA * SROW + k0 + 1];
        v2f b0, b1;
        b0.x = maskval(k0,     l);        b0.y = maskval(k0 + 1, l);
        b1.x = maskval(k0,     16 + l);   b1.y = maskval(k0 + 1, 16 + l);
        c0 = __builtin_amdgcn_wmma_f32_16x16x4_f32(false, a, false, b0, (short)0, c0, false, false);
        c1 = __builtin_amdgcn_wmma_f32_16x16x4_f32(false, a, false, b1, (short)0, c1, false, false);
    }
    // D layout: vgpr v -> M = v + 8*hf, N = l (tile0) / 16+l (tile1)
#pragma unroll
    for (int v = 0; v < 8; ++v) {
        const int row = p0 + v + 8 * hf;
        agg[row * SROW + l] = c0[v];
        if (l < 8) agg[row * SROW + 16 + l] = c1[v];   // cols 16..23 (pads get 0)
    }
}

// ------------------------------------------------ 6) Q = softmax(-agg - unary); last iter: d_out = logQ (B,K,H,W)
__global__ void crf_update_kernel(const float* __restrict__ agg,
                                  const float* __restrict__ unary,
                                  float* __restrict__ Q,
                                  float* __restrict__ out, int writeLog, int N) {
    int n = blockIdx.x * blockDim.x + threadIdx.x;
    if (n >= N) return;
    const float* s = agg + (size_t)n * SROW;
    const float* u = unary + (size_t)n * SROW;
    float logits[KCLS];
    float m = -3.402823e38f;
#pragma unroll
    for (int k = 0; k < KCLS; ++k) { float v = -s[k] - u[k]; logits[k] = v; m = fmaxf(m, v); }
    float sum = 0.0f;
#pragma unroll
    for (int k = 0; k < KCLS; ++k) sum += expf(logits[k] - m);
    float inv = 1.0f / sum;
    float* q = Q + (size_t)n * SROW;
#pragma unroll
    for (int k = 0; k < KCLS; ++k) q[k] = expf(logits[k] - m) * inv;
    q[21] = q[22] = q[23] = 0.0f;
    if (writeLog) {
        float lse = m + logf(sum);
        int b = n >> HWSH, hw = n & (HWPIX - 1);
        float* o = out + ((size_t)b * KCLS) * HWPIX + hw;
#pragma unroll
        for (int k = 0; k < KCLS; ++k) o[(size_t)k * HWPIX] = logits[k] - lse;
    }
}

// ---------------------------------------------------------------- launcher
extern "C" void kernel_launch(void* const* d_in, const int* in_sizes, int n_in,
                              void* d_out, int out_size, void* d_ws, size_t ws_size,
                              hipStream_t stream) {
    const float* feat      = (const float*)d_in[0];
    const float* img       = (const float*)d_in[1];
    const float* dist_diff = (const float*)d_in[2];
    const int*   msg_node  = (const int*)  d_in[3];
    const float* alpha     = (const float*)d_in[4];
    const float* beta      = (const float*)d_in[5];
    const float* gamma     = (const float*)d_in[6];
    const float* w1        = (const float*)d_in[7];
    const float* w2        = (const float*)d_in[8];
    float* out = (float*)d_out;

    const int N = in_sizes[1] / 3;   // 131072
    const int E = in_sizes[2] / 2;   // 1048576

    float* ws    = (float*)d_ws;
    float* Q     = ws;                          // N*SROW
    float* unary = Q     + (size_t)N * SROW;    // N*SROW
    float* agg   = unary + (size_t)N * SROW;    // N*SROW
    float* kern  = agg   + (size_t)N * SROW;    // E

    const int T = 256;

    crf_init_kernel<<<(N + T - 1) / T, T, 0, stream>>>(feat, Q, unary, N);
    crf_edge_kern_kernel<<<(E + T - 1) / T, T, 0, stream>>>(img, dist_diff, msg_node,
                                                            alpha, beta, gamma, w1, w2, kern, E);

    const int n4 = (N * SROW) / 4;
    const int zeroBlocks = 3072;
    const int tiles = N / 16;                    // 8192
    const int wmmaBlocks = tiles / (T / 32);     // 8 waves/block -> 1024 blocks (exact)

    for (int it = 0; it < 10; ++it) {
        crf_zero_kernel<<<zeroBlocks, T, 0, stream>>>((float4*)agg, n4);
        crf_scatter_kernel<<<(E + T - 1) / T, T, 0, stream>>>(Q, kern, msg_node, agg, E);
        crf_wmma_mask_kernel<<<wmmaBlocks, T, 0, stream>>>(agg, N);
        crf_update_kernel<<<(N + T - 1) / T, T, 0, stream>>>(agg, unary, Q, out,
                                                             (it == 9) ? 1 : 0, N);
    }
}